// NeuralHMMClassifier_71356586655838
// MI455X (gfx1250) — compile-verified
//
#include <hip/hip_runtime.h>
#include <math.h>

#define NSTATE 64
#define VOCAB  50257
#define HID    512
#define SEQ    8192
#define NCH    128      // chunks
#define LCH    64       // chunk length (NCH*LCH == SEQ)
#define NVT    3142     // ceil(VOCAB/16) vocab tiles
#define PSTR   68       // padded row stride for LDS P (bank-conflict-free WMMA frags)

typedef __attribute__((ext_vector_type(2))) float v2f;
typedef __attribute__((ext_vector_type(8))) float v8f;

// ---------------- workspace layout (float offsets) ----------------
constexpr size_t O_LOGT   = 0;                                   // 64x64 log-transition
constexpr size_t O_TEXP   = O_LOGT   + 4096;                     // exp(logT)
constexpr size_t O_INITLP = O_TEXP   + 4096;                     // 64
constexpr size_t O_LSE    = O_INITLP + 64;                       // 64 per-state logsumexp over vocab
constexpr size_t O_DELTA  = O_LSE    + 64;                       // SEQ per-step max shift
constexpr size_t O_EMIT   = O_DELTA  + SEQ;                      // SEQ*64 log emission at observed token
constexpr size_t O_ETIL   = O_EMIT   + (size_t)SEQ * 64;         // SEQ*64 exp(emit - delta)
constexpr size_t O_M      = O_ETIL   + (size_t)SEQ * 64;         // NCH x 64x64 fwd chunk products
constexpr size_t O_MS     = O_M      + (size_t)NCH * 4096;       // NCH scales
constexpr size_t O_MP     = O_MS     + NCH;                      // NCH x 64x64 bwd chunk products
constexpr size_t O_MPS    = O_MP     + (size_t)NCH * 4096;       // NCH scales
constexpr size_t O_BV     = O_MPS    + NCH;                      // NCH x 64 fwd boundary vectors
constexpr size_t O_BS     = O_BV     + (size_t)NCH * 64;         // NCH
constexpr size_t O_WV     = O_BS     + NCH;                      // NCH x 64 bwd boundary vectors
constexpr size_t O_WSC    = O_WV     + (size_t)NCH * 64;         // NCH
constexpr size_t O_LOGZ   = O_WSC    + NCH;                      // 1
constexpr size_t O_PART   = O_LOGZ   + 1;                        // NVT*64 partial exp-sums

// ---------------- output layout (float offsets) ----------------
constexpr size_t OUT_A  = 0;                                     // log_alpha.T (S,T)
constexpr size_t OUT_B  = (size_t)NSTATE * SEQ;                  // log_beta.T
constexpr size_t OUT_G  = 2 * (size_t)NSTATE * SEQ;              // log_gamma.T
constexpr size_t OUT_XI = 3 * (size_t)NSTATE * SEQ;              // (T-1,S,S)
constexpr size_t OUT_Z  = OUT_XI + (size_t)(SEQ - 1) * 4096;     // scalar

// ================= K1: transition log-softmax, Texp, init log-probs =================
__global__ void k1_setup(const float* __restrict__ qv, const float* __restrict__ trW,
                         const float* __restrict__ trb, const float* __restrict__ initp,
                         float* __restrict__ ws) {
    __shared__ float y[4096];
    __shared__ float ip[64];
    int tid = threadIdx.x;
    for (int r = tid; r < 4096; r += 256) {
        float acc = trb[r];
        const float* wrow = trW + (size_t)r * HID;
        for (int h = 0; h < HID; ++h) acc = fmaf(wrow[h], qv[h], acc);
        y[r] = acc;
    }
    if (tid < 64) ip[tid] = initp[tid];
    __syncthreads();
    if (tid < 64) {
        float m = -1e30f;
        for (int j = 0; j < 64; ++j) m = fmaxf(m, y[tid * 64 + j]);
        float se = 0.f;
        for (int j = 0; j < 64; ++j) se += expf(y[tid * 64 + j] - m);
        float l = m + logf(se);
        for (int j = 0; j < 64; ++j) {
            float v = y[tid * 64 + j] - l;
            ws[O_LOGT + tid * 64 + j] = v;
            ws[O_TEXP + tid * 64 + j] = expf(v);
        }
        float m2 = -1e30f;
        for (int j = 0; j < 64; ++j) m2 = fmaxf(m2, ip[j]);
        float se2 = 0.f;
        for (int j = 0; j < 64; ++j) se2 += expf(ip[j] - m2);
        ws[O_INITLP + tid] = ip[tid] - m2 - logf(se2);
    }
}

// ============ K2: emission logits GEMM over full vocab (WMMA f32 16x16x4), ============
// ============     per-(block,state) partial sums of exp(logit)            ============
__global__ void k2_lse(const float* __restrict__ semb, const float* __restrict__ eW,
                       const float* __restrict__ eb, float* __restrict__ ws) {
    __shared__ float bsh[16];
    __shared__ float red[128];
    int v0   = blockIdx.x * 16;
    int tid  = threadIdx.x;
    int lane = tid & 31, wv = tid >> 5;   // wave handles states [16*wv, 16*wv+16)
    int lo   = lane & 15, hi = lane >> 4;
    int arow = v0 + lo; if (arow >= VOCAB) arow = VOCAB - 1;
    const float* Ab = eW   + (size_t)arow * HID;           // A: emis_W rows (M=vocab tile)
    const float* Bb = semb + (size_t)(wv * 16 + lo) * HID; // B: relu(state_emb)^T (N=states)
    v8f c = {0.f, 0.f, 0.f, 0.f, 0.f, 0.f, 0.f, 0.f};
    for (int k = 0; k < HID; k += 4) {
        v2f a  = *(const v2f*)(Ab + k + 2 * hi);
        v2f br = *(const v2f*)(Bb + k + 2 * hi);
        v2f b; b.x = fmaxf(br.x, 0.f); b.y = fmaxf(br.y, 0.f);
        c = __builtin_amdgcn_wmma_f32_16x16x4_f32(false, a, false, b, (short)0, c, false, false);
    }
    if (tid < 16) { int vv = v0 + tid; bsh[tid] = (vv < VOCAB) ? eb[vv] : 0.f; }
    __syncthreads();
    float part = 0.f;
    for (int k = 0; k < 8; ++k) {
        int row = k + 8 * hi;                 // D layout: row = k + 8*(lane>=16), col = lane%16
        if (v0 + row < VOCAB) part += expf(c[k] + bsh[row]);
    }
    red[tid] = part;
    __syncthreads();
    if (lane < 16) {
        float tot = red[tid] + red[tid + 16];     // fold the two row-groups (same column)
        ws[O_PART + (size_t)blockIdx.x * 64 + wv * 16 + lo] = tot;
    }
}

// ============ K2b: deterministic fixed-order reduction -> lse[s] ============
__global__ void k2b_reduce(float* __restrict__ ws) {
    int s = threadIdx.x;  // 64 threads
    float acc = 0.f;
    for (int b = 0; b < NVT; ++b) acc += ws[O_PART + (size_t)b * 64 + s];
    ws[O_LSE + s] = logf(acc);
}

// ============ K3: gathered emission GEMM (WMMA), emit log-probs + shifts ============
__global__ void k3_emit(const float* __restrict__ semb, const float* __restrict__ eW,
                        const float* __restrict__ eb, const int* __restrict__ ids,
                        float* __restrict__ ws) {
    __shared__ int   idsh[16];
    __shared__ float lsesh[64];
    __shared__ float em[16 * 64];
    __shared__ float dm[16];
    int t0 = blockIdx.x * 16;
    int tid = threadIdx.x;
    if (tid < 16) idsh[tid] = ids[t0 + tid];
    if (tid < 64) lsesh[tid] = ws[O_LSE + tid];
    __syncthreads();
    int lane = tid & 31, wv = tid >> 5, lo = lane & 15, hi = lane >> 4;
    const float* Ab = eW   + (size_t)idsh[lo] * HID;       // gathered vocab rows (M=16 tokens)
    const float* Bb = semb + (size_t)(wv * 16 + lo) * HID;
    v8f c = {0.f, 0.f, 0.f, 0.f, 0.f, 0.f, 0.f, 0.f};
    for (int k = 0; k < HID; k += 4) {
        v2f a  = *(const v2f*)(Ab + k + 2 * hi);
        v2f br = *(const v2f*)(Bb + k + 2 * hi);
        v2f b; b.x = fmaxf(br.x, 0.f); b.y = fmaxf(br.y, 0.f);
        c = __builtin_amdgcn_wmma_f32_16x16x4_f32(false, a, false, b, (short)0, c, false, false);
    }
    int col = wv * 16 + lo;
    for (int k = 0; k < 8; ++k) {
        int row = k + 8 * hi;
        em[row * 64 + col] = c[k] + eb[idsh[row]] - lsesh[col];
    }
    __syncthreads();
    if (tid < 16) {
        float m = -1e30f;
        for (int j = 0; j < 64; ++j) m = fmaxf(m, em[tid * 64 + j]);
        dm[tid] = m;
        ws[O_DELTA + t0 + tid] = m;
    }
    __syncthreads();
    for (int e = tid; e < 1024; e += 128) {
        int row = e >> 6, s = e & 63;
        float v = em[e];
        ws[O_EMIT + (size_t)(t0 + row) * 64 + s] = v;
        ws[O_ETIL + (size_t)(t0 + row) * 64 + s] = expf(v - dm[row]);
    }
}

// ============ K4: per-chunk products of step matrices A_t = Texp * diag(etil_t) ======
// WMMA version: P lives in LDS (padded stride 68 -> conflict-free b64 A-fragment loads);
// Texp B-fragments preloaded into registers (loop-invariant). 8 waves: wave w owns
// output i-tile (w>>1) x j-tiles {2*(w&1), 2*(w&1)+1}; per step 32 v_wmma + 16 ds_load.
// blocks [0,127): fwd M_c over t in [64c+1, 64c+64]; blocks [127,254): bwd M'_c over
// t in [64c, 64c+63], c = b-126.  Scales carry rescale factors + sum of deltas.
__global__ void k4_chunks(float* __restrict__ ws) {
    __shared__ float P[64 * PSTR];
    __shared__ float red[256];
    int b = blockIdx.x, tid = threadIdx.x;
    int lane = tid & 31, wv = tid >> 5;        // 8 waves
    int lo = lane & 15, hi = lane >> 4;
    bool fwd = (b < (NCH - 1));
    int c = fwd ? b : (b - (NCH - 2));         // bwd: c in [1, NCH-1]
    int tstart = fwd ? (c * LCH + 1) : (c * LCH);
    float* dst    = ws + (fwd ? (O_M  + (size_t)c * 4096) : (O_MP  + (size_t)c * 4096));
    float* dscale = ws + (fwd ? (O_MS + c)                : (O_MPS + c));

    int it  = wv >> 1;                         // i-tile 0..3
    int jt0 = (wv & 1) * 2;                    // first of two j-tiles
    int colA = jt0 * 16 + lo;
    int colB = colA + 16;

    // preload loop-invariant Texp B-fragments into registers (2 tiles x 16 ksteps x 2)
    float btx[2][16][2];
    const float* TexpG = ws + O_TEXP;
#pragma unroll
    for (int jj = 0; jj < 2; ++jj) {
        int col = (jt0 + jj) * 16 + lo;
#pragma unroll
        for (int ks = 0; ks < 16; ++ks) {
            btx[jj][ks][0] = TexpG[(4 * ks + 2 * hi) * 64 + col];
            btx[jj][ks][1] = TexpG[(4 * ks + 2 * hi + 1) * 64 + col];
        }
    }
    for (int x = tid; x < 4096; x += 256)
        P[(x >> 6) * PSTR + (x & 63)] = ((x >> 6) == (x & 63)) ? 1.f : 0.f;
    __syncthreads();

    float lsc = 0.f;
    for (int s = 0; s < LCH; ++s) {
        int t = tstart + s;
        lsc += ws[O_DELTA + t];
        float e0 = ws[O_ETIL + (size_t)t * 64 + colA];
        float e1 = ws[O_ETIL + (size_t)t * 64 + colB];
        v8f c0 = {0.f, 0.f, 0.f, 0.f, 0.f, 0.f, 0.f, 0.f};
        v8f c1 = {0.f, 0.f, 0.f, 0.f, 0.f, 0.f, 0.f, 0.f};
#pragma unroll
        for (int ks = 0; ks < 16; ++ks) {
            // A fragment: P rows it*16+lo, cols 4ks+2hi .. +1 (b64, conflict-free)
            v2f a = *(const v2f*)&P[(it * 16 + lo) * PSTR + 4 * ks + 2 * hi];
            v2f b0 = { btx[0][ks][0], btx[0][ks][1] };
            v2f b1 = { btx[1][ks][0], btx[1][ks][1] };
            c0 = __builtin_amdgcn_wmma_f32_16x16x4_f32(false, a, false, b0, (short)0, c0, false, false);
            c1 = __builtin_amdgcn_wmma_f32_16x16x4_f32(false, a, false, b1, (short)0, c1, false, false);
        }
        __syncthreads();
#pragma unroll
        for (int k = 0; k < 8; ++k) {          // D: row = k + 8*hi, col = lane%16
            int row = it * 16 + k + 8 * hi;
            P[row * PSTR + colA] = c0[k] * e0;
            P[row * PSTR + colB] = c1[k] * e1;
        }
        __syncthreads();
        if ((s & 7) == 7) {                    // periodic rescale to avoid underflow
            float m = 0.f;
            for (int x = tid; x < 4096; x += 256)
                m = fmaxf(m, P[(x >> 6) * PSTR + (x & 63)]);
            red[tid] = m;
            __syncthreads();
            for (int off = 128; off; off >>= 1) {
                if (tid < off) red[tid] = fmaxf(red[tid], red[tid + off]);
                __syncthreads();
            }
            float mm = fmaxf(red[0], 1e-30f);
            float inv = 1.f / mm;
            for (int x = tid; x < 4096; x += 256)
                P[(x >> 6) * PSTR + (x & 63)] *= inv;
            lsc += logf(mm);
            __syncthreads();
        }
    }
    for (int x = tid; x < 4096; x += 256)
        dst[x] = P[(x >> 6) * PSTR + (x & 63)];
    if (tid == 0) *dscale = lsc;
}

// ============ K5: boundary scans (fwd alpha boundaries, bwd beta boundaries) =========
__global__ void k5_scan(float* __restrict__ ws) {
    __shared__ float u[64], un[64], w[64], wn[64];
    int j = threadIdx.x;  // 64 threads
    float a0 = ws[O_INITLP + j] + ws[O_EMIT + j];
    un[j] = a0;
    __syncthreads();
    float m = -1e30f;
    for (int k = 0; k < 64; ++k) m = fmaxf(m, un[k]);
    u[j] = expf(a0 - m);
    float LS = m;
    __syncthreads();
    float s0 = 0.f;
    for (int k = 0; k < 64; ++k) s0 += u[k];
    u[j] /= s0; LS += logf(s0);
    __syncthreads();
    for (int c = 0; c < NCH; ++c) {
        ws[O_BV + (size_t)c * 64 + j] = u[j];
        if (j == 0) ws[O_BS + c] = LS;
        if (c < NCH - 1) {
            const float* M = ws + O_M + (size_t)c * 4096;
            float acc = 0.f;
            for (int i = 0; i < 64; ++i) acc = fmaf(u[i], M[i * 64 + j], acc);
            __syncthreads();
            un[j] = acc;
            __syncthreads();
            float s2 = 0.f;
            for (int k = 0; k < 64; ++k) s2 += un[k];
            u[j] = acc / s2;
            LS += logf(s2) + ws[O_MS + c];
            __syncthreads();
        }
    }
    w[j] = 1.f / 64.f;
    float LS2 = logf(64.f);
    __syncthreads();
    for (int c = NCH - 1; c >= 0; --c) {
        ws[O_WV + (size_t)c * 64 + j] = w[j];
        if (j == 0) ws[O_WSC + c] = LS2;
        if (c > 0) {
            const float* M = ws + O_MP + (size_t)c * 4096;
            float acc = 0.f;
            for (int jj = 0; jj < 64; ++jj) acc = fmaf(M[j * 64 + jj], w[jj], acc);
            __syncthreads();
            wn[j] = acc;
            __syncthreads();
            float s2 = 0.f;
            for (int k = 0; k < 64; ++k) s2 += wn[k];
            w[j] = acc / s2;
            LS2 += logf(s2) + ws[O_MPS + c];
            __syncthreads();
        }
    }
}

// ============ K6: replay chunks forward, write log_alpha (S,T) =============
__global__ void k6_alpha(const float* __restrict__ ws, float* __restrict__ outA) {
    __shared__ float TX[4096];
    __shared__ float p[64], pn[64];
    __shared__ float la[64 * 65];
    int c = blockIdx.x, j = threadIdx.x;  // 64 threads
    for (int x = j; x < 4096; x += 64) TX[x] = ws[O_TEXP + x];
    p[j] = ws[O_BV + (size_t)c * 64 + j];
    float LS = ws[O_BS + c];
    __syncthreads();
    la[j] = logf(p[j]) + LS;
    __syncthreads();
    for (int tl = 1; tl < LCH; ++tl) {
        int t = c * LCH + tl;
        float acc = 0.f;
        for (int i = 0; i < 64; ++i) acc = fmaf(p[i], TX[i * 64 + j], acc);
        acc *= ws[O_ETIL + (size_t)t * 64 + j];
        __syncthreads();
        pn[j] = acc;
        __syncthreads();
        float s = 0.f;
        for (int k = 0; k < 64; ++k) s += pn[k];
        p[j] = acc / s;
        LS += ws[O_DELTA + t] + logf(s);
        la[tl * 65 + j] = logf(p[j]) + LS;
        __syncthreads();
    }
    for (int row = 0; row < 64; ++row)  // coalesced transposed store
        outA[(size_t)row * SEQ + c * LCH + j] = la[j * 65 + row];
}

// ============ K7: replay chunks backward, write log_beta (S,T) =============
__global__ void k7_beta(const float* __restrict__ ws, float* __restrict__ outB) {
    __shared__ float TXt[64 * 65];  // TXt[k*65+i] = Texp[i][k]
    __shared__ float q[64], r[64], qn[64];
    __shared__ float lb[64 * 65];
    int c = blockIdx.x, j = threadIdx.x;  // 64 threads
    for (int x = j; x < 4096; x += 64) TXt[(x & 63) * 65 + (x >> 6)] = ws[O_TEXP + x];
    q[j] = ws[O_WV + (size_t)c * 64 + j];
    float LS = ws[O_WSC + c];
    __syncthreads();
    lb[(LCH - 1) * 65 + j] = logf(q[j]) + LS;
    __syncthreads();
    for (int tl = LCH - 2; tl >= 0; --tl) {
        int t1 = c * LCH + tl + 1;
        r[j] = ws[O_ETIL + (size_t)t1 * 64 + j] * q[j];
        __syncthreads();
        float acc = 0.f;
        for (int k = 0; k < 64; ++k) acc = fmaf(TXt[k * 65 + j], r[k], acc);
        qn[j] = acc;
        __syncthreads();
        float s = 0.f;
        for (int k = 0; k < 64; ++k) s += qn[k];
        q[j] = acc / s;
        LS += ws[O_DELTA + t1] + logf(s);
        lb[tl * 65 + j] = logf(q[j]) + LS;
        __syncthreads();
    }
    for (int row = 0; row < 64; ++row)
        outB[(size_t)row * SEQ + c * LCH + j] = lb[j * 65 + row];
}

// ============ K8: logZ = logsumexp(log_alpha[:, T-1]) =============
__global__ void k8_logz(const float* __restrict__ outA, float* __restrict__ ws,
                        float* __restrict__ outZ) {
    __shared__ float v[64];
    int j = threadIdx.x;
    v[j] = outA[(size_t)j * SEQ + (SEQ - 1)];
    __syncthreads();
    float m = -1e30f;
    for (int k = 0; k < 64; ++k) m = fmaxf(m, v[k]);
    float se = 0.f;
    for (int k = 0; k < 64; ++k) se += expf(v[k] - m);
    if (j == 0) { float lz = m + logf(se); ws[O_LOGZ] = lz; *outZ = lz; }
}

// ============ K9: log_gamma = log_alpha + log_beta - logZ =============
__global__ void k9_gamma(const float* __restrict__ outA, const float* __restrict__ outB,
                         const float* __restrict__ ws, float* __restrict__ outG) {
    size_t idx = (size_t)blockIdx.x * 256 + threadIdx.x;
    if (idx < (size_t)NSTATE * SEQ) outG[idx] = outA[idx] + outB[idx] - ws[O_LOGZ];
}

// ============ K10: log_xi[t,i,j] (bandwidth-bound, LDS-staged operands) =============
__global__ void k10_xi(const float* __restrict__ outA, const float* __restrict__ outB,
                       const float* __restrict__ ws, float* __restrict__ outXi) {
    __shared__ float lt[4096];
    __shared__ float ac[64], bc[64], er[64];
    int t = blockIdx.x, tid = threadIdx.x;
    for (int x = tid; x < 4096; x += 256) lt[x] = ws[O_LOGT + x];
    if (tid < 64) ac[tid] = outA[(size_t)tid * SEQ + t];
    else if (tid < 128) { int s = tid - 64;  bc[s] = outB[(size_t)s * SEQ + t + 1]; }
    else if (tid < 192) { int s = tid - 128; er[s] = ws[O_EMIT + (size_t)(t + 1) * 64 + s]; }
    __syncthreads();
    float lz = ws[O_LOGZ];
    float* dst = outXi + (size_t)t * 4096;
    for (int e = tid; e < 4096; e += 256) {
        int i = e >> 6, jj = e & 63;
        dst[e] = ac[i] + lt[e] + er[jj] + bc[jj] - lz;
    }
}

// =====================================================================
extern "C" void kernel_launch(void* const* d_in, const int* in_sizes, int n_in,
                              void* d_out, int out_size, void* d_ws, size_t ws_size,
                              hipStream_t stream) {
    const float* qv    = (const float*)d_in[0];
    const float* trW   = (const float*)d_in[1];
    const float* trb   = (const float*)d_in[2];
    const float* initp = (const float*)d_in[3];
    const float* semb  = (const float*)d_in[4];
    const float* eW    = (const float*)d_in[5];
    const float* eb    = (const float*)d_in[6];
    const int*   ids   = (const int*)d_in[7];
    float* out = (float*)d_out;
    float* ws  = (float*)d_ws;

    k1_setup  <<<1, 256, 0, stream>>>(qv, trW, trb, initp, ws);
    k2_lse    <<<NVT, 128, 0, stream>>>(semb, eW, eb, ws);
    k2b_reduce<<<1, 64, 0, stream>>>(ws);
    k3_emit   <<<SEQ / 16, 128, 0, stream>>>(semb, eW, eb, ids, ws);
    k4_chunks <<<2 * (NCH - 1), 256, 0, stream>>>(ws);
    k5_scan   <<<1, 64, 0, stream>>>(ws);
    k6_alpha  <<<NCH, 64, 0, stream>>>(ws, out + OUT_A);
    k7_beta   <<<NCH, 64, 0, stream>>>(ws, out + OUT_B);
    k8_logz   <<<1, 64, 0, stream>>>(out + OUT_A, ws, out + OUT_Z);
    k9_gamma  <<<(NSTATE * SEQ + 255) / 256, 256, 0, stream>>>(out + OUT_A, out + OUT_B, ws, out + OUT_G);
    k10_xi    <<<SEQ - 1, 256, 0, stream>>>(out + OUT_A, out + OUT_B, ws, out + OUT_XI);
}